// ContextualAttention_25726854103141
// MI455X (gfx1250) — compile-verified
//
#include <hip/hip_runtime.h>

// ---------------------------------------------------------------------------
// Contextual attention, MI455X (gfx1250, wave32, WMMA).
// All heavy math runs through V_WMMA_F32_16X16X4_F32 (exact fp32 matrix pipe).
// Shapes hardcoded from setup_inputs(): B=4, C=128, H=W=64, rate=2,
// fh=fw=bh=bw=32, P=1024, K_score=1152, K_deconv=1024, K_conv=1152.
//
// LDS layouts chosen so every WMMA operand fragment is ONE aligned b64 load:
//   As[row][k]   : stride 20 floats  -> {A[m][k],A[m][k+1]} contiguous, and
//                  fragment reads are bank-conflict-free (20d = -2 mod 64 has
//                  no solution; 20r mod 64 distinct over r=0..15).
//   Bs pair-interleaved: Bs[(k>>1)*160 + n*2 + (k&1)] -> {B[k][n],B[k+1][n]}
//                  contiguous; +160 = +32 banks puts the two lane-halves on
//                  disjoint bank halves.
// ---------------------------------------------------------------------------

typedef __attribute__((ext_vector_type(2))) float v2f;
typedef __attribute__((ext_vector_type(8))) float v8f;

#define AS_STRIDE 20
#define BS_STRIDE 160
#define AS_ELEMS (64 * AS_STRIDE)
#define BS_ELEMS (8 * BS_STRIDE)

// nearest grid-sample index (align_corners=False, zeros padding), 64 -> 32
__device__ __forceinline__ int ds_index(int o, bool& ok) {
  float g = (float)o / 31.0f * 2.0f - 1.0f;       // linspace(-1,1,32)
  float t = ((g + 1.0f) * 64.0f - 1.0f) * 0.5f;   // ((g+1)*H-1)/2
  float r = rintf(t);                              // round-half-to-even
  ok = (r >= 0.0f) && (r <= 63.0f);
  int i = (int)r;
  i = i < 0 ? 0 : (i > 63 ? 63 : i);
  return i;
}

// ---------------------------------------------------------------------------
// K1: downsample f and b to 32x32 (B,C,32,32)
// ---------------------------------------------------------------------------
__global__ __launch_bounds__(256) void k_downsample(
    const float* __restrict__ f, const float* __restrict__ bfull,
    float* __restrict__ fds, float* __restrict__ bds) {
  int idx = blockIdx.x * 256 + threadIdx.x;      // B*C*1024 = 524288
  if (idx >= 4 * 128 * 1024) return;
  int ox = idx & 31, oy = (idx >> 5) & 31, bc = idx >> 10;
  bool oky, okx;
  int iy = ds_index(oy, oky);
  int ix = ds_index(ox, okx);
  bool ok = oky && okx;
  int s = (bc * 64 + iy) * 64 + ix;
  fds[idx] = ok ? f[s] : 0.0f;
  bds[idx] = ok ? bfull[s] : 0.0f;
}

// ---------------------------------------------------------------------------
// K2: mask downsample + patch-mean -> mm scalar ((mean==0) ? 1 : 0)
// ---------------------------------------------------------------------------
__global__ __launch_bounds__(1024) void k_mask_mm(
    const float* __restrict__ mask, float* __restrict__ mmp) {
  __shared__ float md[1024];
  __shared__ float red[1024];
  const int t = threadIdx.x;
  const int oy = t >> 5, ox = t & 31;
  bool oky, okx;
  int iy = ds_index(oy, oky);
  int ix = ds_index(ox, okx);
  md[t] = (oky && okx) ? mask[iy * 64 + ix] : 0.0f;
  __syncthreads();
  float s = 0.0f;                                 // 3x3 patch at (oy,ox), pad 1
#pragma unroll
  for (int ky = 0; ky < 3; ++ky) {
    int yy = oy - 1 + ky;
    if ((unsigned)yy < 32u) {
#pragma unroll
      for (int kx = 0; kx < 3; ++kx) {
        int xx = ox - 1 + kx;
        if ((unsigned)xx < 32u) s += md[yy * 32 + xx];
      }
    }
  }
  red[t] = s;
  __syncthreads();
  for (int off = 512; off > 0; off >>= 1) {
    if (t < off) red[t] += red[t + off];
    __syncthreads();
  }
  if (t == 0) mmp[0] = (red[0] == 0.0f) ? 1.0f : 0.0f;
}

// ---------------------------------------------------------------------------
// K3: per-c-slot patch norm (the reference's scrambled reshape: for slot c,
// sum over all p of patch element (c0,hp,wp) with L=p*128+c).
// invn[b*128+c] = 1/max(sqrt(sum),1e-4)
// ---------------------------------------------------------------------------
__global__ __launch_bounds__(256) void k_norm(
    const float* __restrict__ bds, float* __restrict__ invn) {
  const int b = blockIdx.x >> 7, c = blockIdx.x & 127;
  const int tid = threadIdx.x;
  float s = 0.0f;
  for (int p = tid; p < 1024; p += 256) {
    int L = p * 128 + c;
    int c0 = L >> 10, hp = (L >> 5) & 31, wp = L & 31;
    const float* base = bds + (b * 128 + c0) * 1024;
#pragma unroll
    for (int ky = 0; ky < 3; ++ky) {
      int yy = hp - 1 + ky;
      if ((unsigned)yy < 32u) {
#pragma unroll
        for (int kx = 0; kx < 3; ++kx) {
          int xx = wp - 1 + kx;
          if ((unsigned)xx < 32u) {
            float v = base[yy * 32 + xx];
            s += v * v;
          }
        }
      }
    }
  }
  __shared__ float red[256];
  red[tid] = s;
  __syncthreads();
  for (int off = 128; off > 0; off >>= 1) {
    if (tid < off) red[tid] += red[tid + off];
    __syncthreads();
  }
  if (tid == 0) invn[blockIdx.x] = 1.0f / fmaxf(sqrtf(red[0]), 1e-4f);
}

// ---------------------------------------------------------------------------
// Shared WMMA micro-tile: one 16-deep K chunk, 4 waves each own 16 M-rows,
// 4 N-tiles of 16 => 16x v_wmma_f32_16x16x4_f32 per wave per chunk.
// A fragment: lane L holds A[M=L&15][2*(L>>4)+{0,1}]  (ISA 16x4 f32 layout)
// B fragment: lane L holds B[2*(L>>4)+{0,1}][N=L&15]
// D fragment: VGPR r -> D[M=r+8*(L>>4)][N=L&15]
// ---------------------------------------------------------------------------
__device__ __forceinline__ void wmma_tile16(const float* As, const float* Bs,
                                            int w16, int lr, int lh,
                                            v8f acc[4]) {
#pragma unroll
  for (int ks = 0; ks < 4; ++ks) {
    const int kb = 4 * ks;
    v2f a = *(const v2f*)(As + (w16 + lr) * AS_STRIDE + kb + 2 * lh);
    const float* brow = Bs + (2 * ks + lh) * BS_STRIDE + lr * 2;
#pragma unroll
    for (int nt = 0; nt < 4; ++nt) {
      v2f bb = *(const v2f*)(brow + nt * 32);
      acc[nt] = __builtin_amdgcn_wmma_f32_16x16x4_f32(
          false, a, false, bb, (short)0, acc[nt], false, false);
    }
  }
}

// ---------------------------------------------------------------------------
// K4: score GEMM. yi[b,p,q] = sum_{c,t} wpatch[b,p,c,t]*invn[b,c] * fpatch[b,c,q,t]
// M=p (1024), N=q=uv (1024), K=c*9+t (1152). Implicit im2col in LDS.
// ---------------------------------------------------------------------------
__global__ __launch_bounds__(128) void k_gemm_scores(
    const float* __restrict__ fds, const float* __restrict__ bds,
    const float* __restrict__ invn, float* __restrict__ outS) {
  __shared__ __align__(16) float As[AS_ELEMS];
  __shared__ __align__(16) float Bs[BS_ELEMS];
  const int b = blockIdx.x, m0 = blockIdx.y * 64, n0 = blockIdx.z * 64;
  const int tid = threadIdx.x, lane = tid & 31, w = tid >> 5;
  const int lr = lane & 15, lh = lane >> 4, w16 = w * 16;
  v8f acc[4] = {};
  for (int k0 = 0; k0 < 1152; k0 += 16) {
#pragma unroll
    for (int i = 0; i < 8; ++i) {                // A tile: 64 rows x 16 k
      int t = i * 128 + tid;
      int row = t >> 4, kc = t & 15;
      int kg = k0 + kc;
      int c = kg / 9, t9 = kg - c * 9;
      int ky = t9 / 3, kx = t9 - ky * 3;
      int L = (m0 + row) * 128 + c;              // scrambled reshape mapping
      int c0 = L >> 10, hp = (L >> 5) & 31, wp = L & 31;
      int yy = hp - 1 + ky, xx = wp - 1 + kx;
      float v = 0.0f;
      if ((unsigned)yy < 32u && (unsigned)xx < 32u)
        v = bds[((b * 128 + c0) * 32 + yy) * 32 + xx];
      As[row * AS_STRIDE + kc] = v;
    }
#pragma unroll
    for (int i = 0; i < 8; ++i) {                // B tile: 16 k x 64 n
      int t = i * 128 + tid;
      int kr = t >> 6, nc = t & 63;
      int kg = k0 + kr;
      int c = kg / 9, t9 = kg - c * 9;
      int ky = t9 / 3, kx = t9 - ky * 3;
      int q = n0 + nc;
      int uy = q >> 5, ux = q & 31;
      int yy = uy - 1 + ky, xx = ux - 1 + kx;
      float v = 0.0f;
      if ((unsigned)yy < 32u && (unsigned)xx < 32u)
        v = fds[((b * 128 + c) * 32 + yy) * 32 + xx] * invn[b * 128 + c];
      Bs[(kr >> 1) * BS_STRIDE + nc * 2 + (kr & 1)] = v;
    }
    __syncthreads();
    wmma_tile16(As, Bs, w16, lr, lh, acc);
    __syncthreads();
  }
#pragma unroll
  for (int nt = 0; nt < 4; ++nt)
#pragma unroll
    for (int r = 0; r < 8; ++r) {
      int m = m0 + w16 + r + 8 * lh;
      int n = n0 + nt * 16 + lr;
      outS[((b << 10) + m) * 1024 + n] = acc[nt][r];
    }
}

// ---------------------------------------------------------------------------
// K5/K6: fuse_diag passes over the (p,q) 1024x1024 score matrix.
// fuse1: out[i,j] = sum_d in[i+d,j+d]; fuse2 does the same on the
// within-32x32-transposed indices (replicating reshape/transpose/reshape).
// ---------------------------------------------------------------------------
__global__ __launch_bounds__(256) void k_fuse1(const float* __restrict__ in,
                                               float* __restrict__ outp) {
  int idx = blockIdx.x * 256 + threadIdx.x;      // 4*1024*1024
  int j = idx & 1023, i = (idx >> 10) & 1023, b = idx >> 20;
  const float* base = in + (b << 20);
  float s = 0.0f;
#pragma unroll
  for (int d = -1; d <= 1; ++d) {
    int ii = i + d, jj = j + d;
    if ((unsigned)ii < 1024u && (unsigned)jj < 1024u) s += base[ii * 1024 + jj];
  }
  outp[idx] = s;
}

__global__ __launch_bounds__(256) void k_fuse2(const float* __restrict__ in,
                                               float* __restrict__ outp) {
  int idx = blockIdx.x * 256 + threadIdx.x;
  int j = idx & 1023, i = (idx >> 10) & 1023, b = idx >> 20;
  int pp = ((i & 31) << 5) | (i >> 5);           // T(i): swap (hi5,lo5)
  int qq = ((j & 31) << 5) | (j >> 5);
  const float* base = in + (b << 20);
  float s = 0.0f;
#pragma unroll
  for (int d = -1; d <= 1; ++d) {
    int i2 = pp + d, j2 = qq + d;
    if ((unsigned)i2 < 1024u && (unsigned)j2 < 1024u) {
      int ti = ((i2 & 31) << 5) | (i2 >> 5);
      int tj = ((j2 & 31) << 5) | (j2 >> 5);
      s += base[ti * 1024 + tj];
    }
  }
  outp[idx] = s;
}

// ---------------------------------------------------------------------------
// K7: softmax over p (dim with stride 1024) for each (b,q); logits = x*mm*10,
// output scaled by mm. One block: 32 q-columns x 8 row-lanes.
// ---------------------------------------------------------------------------
__global__ __launch_bounds__(256) void k_softmax(float* __restrict__ S,
                                                 const float* __restrict__ mmp) {
  const int b = blockIdx.x >> 5;
  const int tx = threadIdx.x & 31, ty = threadIdx.x >> 5;
  const int q = ((blockIdx.x & 31) << 5) + tx;
  const float m = mmp[0];
  const float sc = m * 10.0f;
  float* col = S + (b << 20) + q;
  __shared__ float red[8][32];
  float mx = -3.402823e38f;
  for (int p = ty; p < 1024; p += 8) mx = fmaxf(mx, col[p << 10] * sc);
  red[ty][tx] = mx;
  __syncthreads();
  if (ty == 0) {
    float v = red[0][tx];
#pragma unroll
    for (int r = 1; r < 8; ++r) v = fmaxf(v, red[r][tx]);
    red[0][tx] = v;
  }
  __syncthreads();
  const float M = red[0][tx];
  __syncthreads();
  float sm = 0.0f;
  for (int p = ty; p < 1024; p += 8) sm += expf(col[p << 10] * sc - M);
  red[ty][tx] = sm;
  __syncthreads();
  if (ty == 0) {
    float v = 0.0f;
#pragma unroll
    for (int r = 0; r < 8; ++r) v += red[r][tx];
    red[0][tx] = v;
  }
  __syncthreads();
  const float inv = m / red[0][tx];
  for (int p = ty; p < 1024; p += 8)
    col[p << 10] = expf(col[p << 10] * sc - M) * inv;
}

// ---------------------------------------------------------------------------
// K8: transposed-conv as GEMM: T[b, m=(c*16+ky*4+kx), q=(u,v)] =
//   sum_p raw_w[b,p,c,ky,kx] * soft[b,p,q]   (raw_w via scrambled mapping)
// ---------------------------------------------------------------------------
__global__ __launch_bounds__(128) void k_gemm_deconv(
    const float* __restrict__ bfull, const float* __restrict__ S,
    float* __restrict__ T) {
  __shared__ __align__(16) float As[AS_ELEMS];
  __shared__ __align__(16) float Bs[BS_ELEMS];
  const int b = blockIdx.x, m0 = blockIdx.y * 64, n0 = blockIdx.z * 64;
  const int tid = threadIdx.x, lane = tid & 31, w = tid >> 5;
  const int lr = lane & 15, lh = lane >> 4, w16 = w * 16;
  v8f acc[4] = {};
  for (int k0 = 0; k0 < 1024; k0 += 16) {
#pragma unroll
    for (int i = 0; i < 8; ++i) {                // A: raw_w gather
      int t = i * 128 + tid;
      int row = t >> 4, kc = t & 15;
      int m = m0 + row;
      int c = m >> 4, kk = m & 15, ky = kk >> 2, kx = kk & 3;
      int L = (k0 + kc) * 128 + c;               // p*128+c scramble
      int c0 = L >> 10, hp = (L >> 5) & 31, wp = L & 31;
      int yy = 2 * hp - 1 + ky, xx = 2 * wp - 1 + kx;
      float v = 0.0f;
      if ((unsigned)yy < 64u && (unsigned)xx < 64u)
        v = bfull[((b * 128 + c0) * 64 + yy) * 64 + xx];
      As[row * AS_STRIDE + kc] = v;
    }
#pragma unroll
    for (int i = 0; i < 8; ++i) {                // B: softmaxed scores, dense
      int t = i * 128 + tid;
      int kr = t >> 6, nc = t & 63;
      float v = S[((b << 10) + k0 + kr) * 1024 + n0 + nc];
      Bs[(kr >> 1) * BS_STRIDE + nc * 2 + (kr & 1)] = v;
    }
    __syncthreads();
    wmma_tile16(As, Bs, w16, lr, lh, acc);
    __syncthreads();
  }
#pragma unroll
  for (int nt = 0; nt < 4; ++nt)
#pragma unroll
    for (int r = 0; r < 8; ++r) {
      int m = m0 + w16 + r + 8 * lh;
      int n = n0 + nt * 16 + lr;
      T[((b << 11) + m) * 1024 + n] = acc[nt][r];
    }
}

// ---------------------------------------------------------------------------
// K9: deterministic 4-tap gather of the stride-2 scatter: yout = 0.25*sum
// ---------------------------------------------------------------------------
__global__ __launch_bounds__(256) void k_gather(const float* __restrict__ T,
                                                float* __restrict__ yout) {
  int idx = blockIdx.x * 256 + threadIdx.x;      // B*C*64*64
  int x = idx & 63, y = (idx >> 6) & 63, c = (idx >> 12) & 127, b = idx >> 19;
  float s = 0.0f;
  for (int ky = (y + 1) & 1; ky < 4; ky += 2) {
    int u = (y + 1 - ky) >> 1;
    if ((unsigned)u < 32u) {
      for (int kx = (x + 1) & 1; kx < 4; kx += 2) {
        int v = (x + 1 - kx) >> 1;
        if ((unsigned)v < 32u)
          s += T[((b << 11) + (c * 16 + ky * 4 + kx)) * 1024 + u * 32 + v];
      }
    }
  }
  yout[idx] = 0.25f * s;
}

// ---------------------------------------------------------------------------
// K10: 3x3 conv (C=128->128, 64x64, pad 1) as implicit GEMM + bias.
// M=co(128), N=y*64+x(4096), K=ci*9+t(1152).
// ---------------------------------------------------------------------------
__global__ __launch_bounds__(128) void k_gemm_conv(
    const float* __restrict__ in, const float* __restrict__ wgt,
    const float* __restrict__ bias, float* __restrict__ outp) {
  __shared__ __align__(16) float As[AS_ELEMS];
  __shared__ __align__(16) float Bs[BS_ELEMS];
  const int b = blockIdx.x, m0 = blockIdx.y * 64, n0 = blockIdx.z * 64;
  const int tid = threadIdx.x, lane = tid & 31, w = tid >> 5;
  const int lr = lane & 15, lh = lane >> 4, w16 = w * 16;
  v8f acc[4] = {};
  for (int k0 = 0; k0 < 1152; k0 += 16) {
#pragma unroll
    for (int i = 0; i < 8; ++i) {                // A: weights (flat co*1152+k)
      int t = i * 128 + tid;
      int row = t >> 4, kc = t & 15;
      As[row * AS_STRIDE + kc] = wgt[(m0 + row) * 1152 + k0 + kc];
    }
#pragma unroll
    for (int i = 0; i < 8; ++i) {                // B: input im2col
      int t = i * 128 + tid;
      int kr = t >> 6, nc = t & 63;
      int kg = k0 + kr;
      int ci = kg / 9, t9 = kg - ci * 9;
      int ky = t9 / 3, kx = t9 - ky * 3;
      int n = n0 + nc;
      int yy = (n >> 6) - 1 + ky, xx = (n & 63) - 1 + kx;
      float v = 0.0f;
      if ((unsigned)yy < 64u && (unsigned)xx < 64u)
        v = in[((b * 128 + ci) * 64 + yy) * 64 + xx];
      Bs[(kr >> 1) * BS_STRIDE + nc * 2 + (kr & 1)] = v;
    }
    __syncthreads();
    wmma_tile16(As, Bs, w16, lr, lh, acc);
    __syncthreads();
  }
#pragma unroll
  for (int nt = 0; nt < 4; ++nt)
#pragma unroll
    for (int r = 0; r < 8; ++r) {
      int m = m0 + w16 + r + 8 * lh;
      int n = n0 + nt * 16 + lr;
      outp[(b * 128 + m) * 4096 + n] = acc[nt][r] + bias[m];
    }
}

// ---------------------------------------------------------------------------
// Launch. Workspace arena (bytes): fds@0(2MB) bds@4MB(2MB) invn@8MB mm@8MB+64K
// SA@16MB(16MB) SB/T@34MB(up to 32MB) yout@16MB(reuse, 8MB) c1@24MB(8MB).
// Peak use ~66MB.
// ---------------------------------------------------------------------------
extern "C" void kernel_launch(void* const* d_in, const int* in_sizes, int n_in,
                              void* d_out, int out_size, void* d_ws,
                              size_t ws_size, hipStream_t stream) {
  (void)in_sizes; (void)n_in; (void)out_size; (void)ws_size;
  const float* f     = (const float*)d_in[0];
  const float* bfull = (const float*)d_in[1];
  const float* mask  = (const float*)d_in[2];
  const float* w1    = (const float*)d_in[3];
  const float* b1    = (const float*)d_in[4];
  const float* w2    = (const float*)d_in[5];
  const float* b2    = (const float*)d_in[6];
  float* out = (float*)d_out;
  char* ws = (char*)d_ws;

  float* fds  = (float*)(ws);
  float* bds  = (float*)(ws + (4ull << 20));
  float* invn = (float*)(ws + (8ull << 20));
  float* mm   = (float*)(ws + (8ull << 20) + 65536);
  float* SA   = (float*)(ws + (16ull << 20));   // scores (16 MB)
  float* SB   = (float*)(ws + (34ull << 20));   // fuse temp; later T (32 MB)
  float* T    = SB;
  float* yout = SA;                              // SA dead after deconv GEMM
  float* c1   = (float*)(ws + (24ull << 20));    // also inside dead SA region

  k_downsample<<<2048, 256, 0, stream>>>(f, bfull, fds, bds);
  k_mask_mm<<<1, 1024, 0, stream>>>(mask, mm);
  k_norm<<<512, 256, 0, stream>>>(bds, invn);
  k_gemm_scores<<<dim3(4, 16, 16), 128, 0, stream>>>(fds, bds, invn, SA);
  k_fuse1<<<16384, 256, 0, stream>>>(SA, SB);
  k_fuse2<<<16384, 256, 0, stream>>>(SB, SA);
  k_softmax<<<128, 256, 0, stream>>>(SA, mm);
  k_gemm_deconv<<<dim3(4, 32, 16), 128, 0, stream>>>(bfull, SA, T);
  k_gather<<<8192, 256, 0, stream>>>(T, yout);
  k_gemm_conv<<<dim3(4, 2, 64), 128, 0, stream>>>(yout, w1, b1, c1);
  k_gemm_conv<<<dim3(4, 2, 64), 128, 0, stream>>>(c1, w2, b2, out);
}